// PRGAT_23802708754506
// MI455X (gfx1250) — compile-verified
//
#include <hip/hip_runtime.h>
#include <hip/hip_bf16.h>
#include <math.h>

// ---------------------------------------------------------------------------
// Fused per-batch-block transformer kernel for MI455X (gfx1250, wave32).
// One workgroup (256 threads = 8 waves) processes 8 batches = 16 rows (M=16).
// All GEMMs use V_WMMA_F32_16X16X4_F32 (16x16 tile, K stepped by 4).
// ---------------------------------------------------------------------------

typedef float v8f __attribute__((ext_vector_type(8)));
typedef float v2f __attribute__((ext_vector_type(2)));

#define DD   512
#define LDA  516   // LDS row stride in floats: 512+4 -> conflict-free A-frag loads
#define ROWS 16

__device__ __forceinline__ v8f vzero() {
  v8f c;
#pragma unroll
  for (int i = 0; i < 8; ++i) c[i] = 0.0f;
  return c;
}

__device__ __forceinline__ float wave_sum(float v) {
#pragma unroll
  for (int off = 16; off > 0; off >>= 1) v += __shfl_xor(v, off, 32);
  return v;
}

// C(16x16) += A(16xK, LDS) * W(Kxldw, global)[:, col-tile]
// A-frag layout (ISA 7.12.2, 32-bit A 16x4): lanes 0-15 hold K=k0,k0+1;
// lanes 16-31 hold K=k0+2,k0+3 (contiguous pair -> ds_load_b64).
// B-frag mirrored: VGPR0 = row k0+c, VGPR1 = row k0+c+1 at column `col`.
__device__ __forceinline__ v8f wmma_tile_f32(const float* As, int lda,
                                             const float* __restrict__ W,
                                             int ldw, int col, int K,
                                             int cshift, int m, v8f c) {
#pragma unroll 4
  for (int k0 = 0; k0 < K; k0 += 4) {
    v2f a, b;
    const float* ar = As + m * lda + k0 + cshift;
    a.x = ar[0];
    a.y = ar[1];
    const float* br = W + (size_t)(k0 + cshift) * ldw + col;
    b.x = br[0];
    b.y = br[ldw];
    c = __builtin_amdgcn_wmma_f32_16x16x4_f32(false, a, false, b, (short)0, c,
                                              false, false);
  }
  return c;
}

__global__ __launch_bounds__(256) void prgat_fused_kernel(
    const float* __restrict__ x,
    const float* __restrict__ wq, const float* __restrict__ bq,
    const float* __restrict__ wk, const float* __restrict__ bk,
    const float* __restrict__ wv, const float* __restrict__ bv,
    const float* __restrict__ wo, const float* __restrict__ bo,
    const float* __restrict__ g1, const float* __restrict__ be1,
    const float* __restrict__ wf1, const float* __restrict__ bf1,
    const float* __restrict__ wf2, const float* __restrict__ bf2,
    const float* __restrict__ g2, const float* __restrict__ be2,
    const float* __restrict__ ws1, const float* __restrict__ bs1,
    const float* __restrict__ ws2, const float* __restrict__ bs2,
    float* __restrict__ outFused, float* __restrict__ outAlpha,
    float* __restrict__ outRef) {
  extern __shared__ __align__(16) float sm[];
  float* XS = sm;                  // 16 x 516 : x, later FFN hidden half, later s1
  float* BA = sm + ROWS * LDA;     // Q -> attn_out -> (h+ffn) -> refined
  float* BB = sm + 2 * ROWS * LDA; // K -> V -> h
  float* ATT = sm + 3 * ROWS * LDA; // 8 batches x 2x2 attention weights
  float* SROW = ATT + 32;           // 16 pooling scores
  float* ALPH = SROW + 16;          // 8 batches x 2 pooling alphas

  const int tid = threadIdx.x;
  const int lane = tid & 31;
  const int wave = tid >> 5;
  const int b0 = blockIdx.x * 8; // first batch of this block
  const int m16 = lane & 15;
  const int cshift = (lane < 16) ? 0 : 2;

  // ---- P0: stage x block (16 x 512) into LDS -----------------------------
  const float* xg = x + (size_t)b0 * 2 * DD;
  for (int i = tid; i < ROWS * (DD / 4); i += 256) {
    int r = i >> 7, d4 = (i & 127) << 2;
    const float4 v = *reinterpret_cast<const float4*>(xg + r * DD + d4);
    *reinterpret_cast<float4*>(&XS[r * LDA + d4]) = v;
  }
  __syncthreads();

  // ---- P1: Q -> BA, K -> BB ---------------------------------------------
#pragma unroll
  for (int t = 0; t < 4; ++t) {
    const int j = (wave + t * 8) * 16;
    const int col = j + m16;
    v8f cq = wmma_tile_f32(XS, LDA, wq, DD, col, DD, cshift, m16, vzero());
    v8f ck = wmma_tile_f32(XS, LDA, wk, DD, col, DD, cshift, m16, vzero());
    const float bqc = bq[col], bkc = bk[col];
#pragma unroll
    for (int r = 0; r < 8; ++r) {
      const int row = (lane < 16) ? r : (r + 8);
      BA[row * LDA + col] = cq[r] + bqc;
      BB[row * LDA + col] = ck[r] + bkc;
    }
  }
  __syncthreads();

  // ---- P2: 2x2 attention per batch (wave w -> batch w) -------------------
  {
    const int b = wave, r0 = b * 2;
    float s00 = 0.f, s01 = 0.f, s10 = 0.f, s11 = 0.f;
    for (int d = lane; d < DD; d += 32) {
      const float q0 = BA[r0 * LDA + d], q1 = BA[(r0 + 1) * LDA + d];
      const float k0 = BB[r0 * LDA + d], k1 = BB[(r0 + 1) * LDA + d];
      s00 = fmaf(q0, k0, s00); s01 = fmaf(q0, k1, s01);
      s10 = fmaf(q1, k0, s10); s11 = fmaf(q1, k1, s11);
    }
    s00 = wave_sum(s00); s01 = wave_sum(s01);
    s10 = wave_sum(s10); s11 = wave_sum(s11);
    if (lane == 0) {
      const float inv = 1.0f / (sqrtf((float)DD) + 1e-8f);
      float a[4] = {s00 * inv, s01 * inv, s10 * inv, s11 * inv};
#pragma unroll
      for (int i = 0; i < 4; ++i) a[i] = (a[i] < 0.f) ? 0.2f * a[i] : a[i];
#pragma unroll
      for (int n = 0; n < 2; ++n) {
        const float x0 = a[n * 2], x1 = a[n * 2 + 1];
        const float mx = fmaxf(x0, x1);
        const float e0 = __expf(x0 - mx), e1 = __expf(x1 - mx);
        const float rs = 1.f / (e0 + e1);
        ATT[b * 4 + n * 2 + 0] = e0 * rs;
        ATT[b * 4 + n * 2 + 1] = e1 * rs;
      }
    }
  }
  __syncthreads();

  // ---- P3: V -> BB (overwrites K) ---------------------------------------
#pragma unroll
  for (int t = 0; t < 4; ++t) {
    const int j = (wave + t * 8) * 16;
    const int col = j + m16;
    v8f cv = wmma_tile_f32(XS, LDA, wv, DD, col, DD, cshift, m16, vzero());
    const float bvc = bv[col];
#pragma unroll
    for (int r = 0; r < 8; ++r) {
      const int row = (lane < 16) ? r : (r + 8);
      BB[row * LDA + col] = cv[r] + bvc;
    }
  }
  __syncthreads();

  // ---- P4: attn_out = attn @ V -> BA (overwrites Q) ----------------------
  for (int i = tid; i < ROWS * DD; i += 256) {
    const int m = i >> 9, d = i & (DD - 1);
    const int b = m >> 1, n = m & 1;
    const float a0 = ATT[b * 4 + n * 2 + 0], a1 = ATT[b * 4 + n * 2 + 1];
    BA[m * LDA + d] =
        a0 * BB[(b * 2) * LDA + d] + a1 * BB[(b * 2 + 1) * LDA + d];
  }
  __syncthreads();

  // ---- P5: o = attn_out @ Wo + bo ; t = x + o -> BB ----------------------
#pragma unroll
  for (int t = 0; t < 4; ++t) {
    const int j = (wave + t * 8) * 16;
    const int col = j + m16;
    v8f co = wmma_tile_f32(BA, LDA, wo, DD, col, DD, cshift, m16, vzero());
    const float boc = bo[col];
#pragma unroll
    for (int r = 0; r < 8; ++r) {
      const int row = (lane < 16) ? r : (r + 8);
      BB[row * LDA + col] = co[r] + boc + XS[row * LDA + col];
    }
  }
  __syncthreads();

  // ---- P6: h = LN(t)*g1 + be1 in place (BB) ------------------------------
#pragma unroll
  for (int rr = 0; rr < 2; ++rr) {
    const int row = wave * 2 + rr;
    float s = 0.f, q = 0.f;
    for (int d = lane; d < DD; d += 32) {
      const float v = BB[row * LDA + d];
      s += v; q = fmaf(v, v, q);
    }
    s = wave_sum(s); q = wave_sum(q);
    const float mu = s * (1.f / DD);
    const float rstd = rsqrtf(fmaxf(q * (1.f / DD) - mu * mu, 0.f) + 1e-5f);
    for (int d = lane; d < DD; d += 32) {
      const float v = (BB[row * LDA + d] - mu) * rstd;
      BB[row * LDA + d] = v * g1[d] + be1[d];
    }
  }
  __syncthreads();

  // ---- P7: FFN. hidden (1024) computed in two 512-halves through XS, ----
  //      ffn accumulated in persistent WMMA accumulators.
  v8f acc[4] = {vzero(), vzero(), vzero(), vzero()};
  for (int half = 0; half < 2; ++half) {
#pragma unroll
    for (int t = 0; t < 4; ++t) {
      const int j = (wave + t * 8) * 16;
      const int col = j + m16;
      const int hcol = half * DD + col;
      v8f ch =
          wmma_tile_f32(BB, LDA, wf1, 2 * DD, hcol, DD, cshift, m16, vzero());
      const float bc = bf1[hcol];
#pragma unroll
      for (int r = 0; r < 8; ++r) {
        const int row = (lane < 16) ? r : (r + 8);
        XS[row * LDA + col] = fmaxf(ch[r] + bc, 0.f);
      }
    }
    __syncthreads();
    const float* w2h = wf2 + (size_t)half * DD * DD;
#pragma unroll
    for (int t = 0; t < 4; ++t) {
      const int j = (wave + t * 8) * 16;
      acc[t] =
          wmma_tile_f32(XS, LDA, w2h, DD, j + m16, DD, cshift, m16, acc[t]);
    }
    __syncthreads();
  }
#pragma unroll
  for (int t = 0; t < 4; ++t) {
    const int j = (wave + t * 8) * 16;
    const int col = j + m16;
    const float bc = bf2[col];
#pragma unroll
    for (int r = 0; r < 8; ++r) {
      const int row = (lane < 16) ? r : (r + 8);
      BA[row * LDA + col] = acc[t][r] + bc + BB[row * LDA + col];
    }
  }
  __syncthreads();

  // ---- P8: refined = LN(h+ffn)*g2 + be2 in place (BA) + write to HBM -----
  float* refg = outRef + (size_t)b0 * 2 * DD;
#pragma unroll
  for (int rr = 0; rr < 2; ++rr) {
    const int row = wave * 2 + rr;
    float s = 0.f, q = 0.f;
    for (int d = lane; d < DD; d += 32) {
      const float v = BA[row * LDA + d];
      s += v; q = fmaf(v, v, q);
    }
    s = wave_sum(s); q = wave_sum(q);
    const float mu = s * (1.f / DD);
    const float rstd = rsqrtf(fmaxf(q * (1.f / DD) - mu * mu, 0.f) + 1e-5f);
    for (int d = lane; d < DD; d += 32) {
      const float v = (BA[row * LDA + d] - mu) * rstd * g2[d] + be2[d];
      BA[row * LDA + d] = v;
      refg[row * DD + d] = v;
    }
  }
  __syncthreads();

  // ---- P9: s1 = relu(refined @ Ws1 + bs1) -> XS --------------------------
#pragma unroll
  for (int t = 0; t < 4; ++t) {
    const int j = (wave + t * 8) * 16;
    const int col = j + m16;
    v8f cs = wmma_tile_f32(BA, LDA, ws1, DD, col, DD, cshift, m16, vzero());
    const float bc = bs1[col];
#pragma unroll
    for (int r = 0; r < 8; ++r) {
      const int row = (lane < 16) ? r : (r + 8);
      XS[row * LDA + col] = fmaxf(cs[r] + bc, 0.f);
    }
  }
  __syncthreads();

  // ---- P10: per-row pooling score = s1 . ws2 + bs2 -----------------------
#pragma unroll
  for (int rr = 0; rr < 2; ++rr) {
    const int row = wave * 2 + rr;
    float s = 0.f;
    for (int d = lane; d < DD; d += 32) s = fmaf(XS[row * LDA + d], ws2[d], s);
    s = wave_sum(s);
    if (lane == 0) SROW[row] = s + bs2[0];
  }
  __syncthreads();

  // ---- P11: alpha = softmax over N=2, write to HBM -----------------------
  if (tid < 8) {
    const int b = tid;
    const float x0 = SROW[2 * b], x1 = SROW[2 * b + 1];
    const float mx = fmaxf(x0, x1);
    const float e0 = __expf(x0 - mx), e1 = __expf(x1 - mx);
    const float rs = 1.f / (e0 + e1);
    ALPH[2 * b + 0] = e0 * rs;
    ALPH[2 * b + 1] = e1 * rs;
    outAlpha[(size_t)(b0 + b) * 2 + 0] = e0 * rs;
    outAlpha[(size_t)(b0 + b) * 2 + 1] = e1 * rs;
  }
  __syncthreads();

  // ---- P12: fused = LN_paramfree(alpha-pooled refined) -> HBM ------------
  {
    const int b = wave;
    const float a0 = ALPH[2 * b], a1 = ALPH[2 * b + 1];
    float s = 0.f, q = 0.f;
    for (int d = lane; d < DD; d += 32) {
      const float f =
          a0 * BA[(2 * b) * LDA + d] + a1 * BA[(2 * b + 1) * LDA + d];
      s += f; q = fmaf(f, f, q);
    }
    s = wave_sum(s); q = wave_sum(q);
    const float mu = s * (1.f / DD);
    const float rstd = rsqrtf(fmaxf(q * (1.f / DD) - mu * mu, 0.f) + 1e-5f);
    float* fg = outFused + (size_t)(b0 + b) * DD;
    for (int d = lane; d < DD; d += 32) {
      const float f =
          a0 * BA[(2 * b) * LDA + d] + a1 * BA[(2 * b + 1) * LDA + d];
      fg[d] = (f - mu) * rstd;
    }
  }
}

extern "C" void kernel_launch(void* const* d_in, const int* in_sizes, int n_in,
                              void* d_out, int out_size, void* d_ws,
                              size_t ws_size, hipStream_t stream) {
  const float* x   = (const float*)d_in[0];
  const float* wq  = (const float*)d_in[1];
  const float* bq  = (const float*)d_in[2];
  const float* wk  = (const float*)d_in[3];
  const float* bk  = (const float*)d_in[4];
  const float* wv  = (const float*)d_in[5];
  const float* bv  = (const float*)d_in[6];
  const float* wo  = (const float*)d_in[7];
  const float* bo  = (const float*)d_in[8];
  const float* g1  = (const float*)d_in[9];
  const float* be1 = (const float*)d_in[10];
  const float* wf1 = (const float*)d_in[11];
  const float* bf1 = (const float*)d_in[12];
  const float* wf2 = (const float*)d_in[13];
  const float* bf2 = (const float*)d_in[14];
  const float* g2  = (const float*)d_in[15];
  const float* be2 = (const float*)d_in[16];
  const float* ws1 = (const float*)d_in[17];
  const float* bs1 = (const float*)d_in[18];
  const float* ws2 = (const float*)d_in[19];
  const float* bs2 = (const float*)d_in[20];

  const int B = in_sizes[0] / (2 * DD); // 65536
  float* out = (float*)d_out;
  float* outFused = out;                               // B*D
  float* outAlpha = out + (size_t)B * DD;              // B*2
  float* outRef   = outAlpha + (size_t)B * 2;          // B*2*D

  const size_t shmem = (size_t)(3 * ROWS * LDA + 64) * sizeof(float); // ~99.3 KB
  static_assert(3 * ROWS * LDA + 64 >= 3 * ROWS * LDA + 32 + 16 + 16, "lds");

  hipFuncSetAttribute(reinterpret_cast<const void*>(prgat_fused_kernel),
                      hipFuncAttributeMaxDynamicSharedMemorySize, (int)shmem);

  dim3 grid(B / 8), block(256);
  prgat_fused_kernel<<<grid, block, shmem, stream>>>(
      x, wq, bq, wk, bk, wv, bv, wo, bo, g1, be1, wf1, bf1, wf2, bf2, g2, be2,
      ws1, bs1, ws2, bs2, outFused, outAlpha, outRef);
}